// GraphPairClassifier_51376398795254
// MI455X (gfx1250) — compile-verified
//
#include <hip/hip_runtime.h>
#include <hip/hip_bf16.h>
#include <math.h>

typedef __attribute__((ext_vector_type(2))) float v2f;
typedef __attribute__((ext_vector_type(8))) float v8f;

#define NGRAPHS 1024

// ---------------- degree accumulation (deg[dst] += 1 per edge) ----------------
__global__ void deg_kernel(const int* __restrict__ dst, int E, float* __restrict__ deg) {
    int e = blockIdx.x * 256 + threadIdx.x;
    if (e < E) atomicAdd(&deg[dst[e]], 1.0f);
}

// dinv = rsqrt(deg + 1)   (the +1 is the self-loop; deg >= 1 always afterwards)
__global__ void dinv_kernel(const float* __restrict__ deg, int N, float* __restrict__ dinv) {
    int i = blockIdx.x * 256 + threadIdx.x;
    if (i < N) dinv[i] = rsqrtf(deg[i] + 1.0f);
}

// ---------------- H = X @ W  via V_WMMA_F32_16X16X4_F32 ----------------
// X: [N,32] row-major, W: [32,64] row-major, H: [N,64].
// One wave computes a 16x64 output slab: 4 f32 accumulators, K looped in steps of 4.
__global__ void gemm_xw_wmma(const float* __restrict__ X,
                             const float* __restrict__ W,
                             float* __restrict__ H, int N) {
    const int lane = threadIdx.x & 31;
    const int wv   = threadIdx.x >> 5;
    const int nrt  = (N + 15) >> 4;
    const int rt   = blockIdx.x * 2 + wv;           // row tile (16 rows)
    if (rt >= nrt) return;                          // whole-wave uniform exit

    const int rloc  = lane & 15;                    // row within tile (A), col within tile (B/C/D)
    const int kh    = (lane >> 4) * 2;              // K sub-offset per half-wave
    const int rowA  = rt * 16 + rloc;
    const int rowAc = rowA < N ? rowA : N - 1;

    v8f acc0 = {}, acc1 = {}, acc2 = {}, acc3 = {};

    for (int k0 = 0; k0 < 32; k0 += 4) {
        // A fragment: lane holds X[rowA][k0+kh], X[rowA][k0+kh+1]
        v2f a;
        a.x = X[(size_t)rowAc * 32 + k0 + kh];
        a.y = X[(size_t)rowAc * 32 + k0 + kh + 1];
        // B fragments: lane holds W[k0+kh][col], W[k0+kh+1][col] per 16-col tile
        const float* w0 = W + (size_t)(k0 + kh) * 64;
        const float* w1 = W + (size_t)(k0 + kh + 1) * 64;
        v2f b0, b1, b2, b3;
        b0.x = w0[rloc];      b0.y = w1[rloc];
        b1.x = w0[rloc + 16]; b1.y = w1[rloc + 16];
        b2.x = w0[rloc + 32]; b2.y = w1[rloc + 32];
        b3.x = w0[rloc + 48]; b3.y = w1[rloc + 48];

        acc0 = __builtin_amdgcn_wmma_f32_16x16x4_f32(false, a, false, b0, (short)0, acc0, false, false);
        acc1 = __builtin_amdgcn_wmma_f32_16x16x4_f32(false, a, false, b1, (short)0, acc1, false, false);
        acc2 = __builtin_amdgcn_wmma_f32_16x16x4_f32(false, a, false, b2, (short)0, acc2, false, false);
        acc3 = __builtin_amdgcn_wmma_f32_16x16x4_f32(false, a, false, b3, (short)0, acc3, false, false);
    }

    // C/D layout: VGPR r holds row M = r + 8*(lane>=16), col = lane&15
    const int rbase = (lane >> 4) * 8;
    for (int r = 0; r < 8; ++r) {
        int row = rt * 16 + rbase + r;
        if (row < N) {
            float* o = H + (size_t)row * 64 + rloc;
            o[0]  = acc0[r];
            o[16] = acc1[r];
            o[32] = acc2[r];
            o[48] = acc3[r];
        }
    }
}

// ---------------- edge scatter: agg[dst] += h[src] * dinv[src]*dinv[dst] ----------------
// 64 threads per edge: contiguous 256B gather + contiguous f32 atomics.
__global__ void scatter_kernel(const int* __restrict__ src, const int* __restrict__ dst,
                               const float* __restrict__ dinv,
                               const float* __restrict__ H,
                               float* __restrict__ agg, int E) {
    long long gid = (long long)blockIdx.x * 256 + threadIdx.x;
    int e = (int)(gid >> 6);
    int f = (int)(gid & 63);
    if (e < E) {
        int s = src[e], d = dst[e];
        float nrm = dinv[s] * dinv[d];
        atomicAdd(&agg[(size_t)d * 64 + f], H[(size_t)s * 64 + f] * nrm);
    }
}

// ---------------- self-loop + bias + ReLU + mean-pool accumulation ----------------
__global__ void pool_kernel(const float* __restrict__ agg, const float* __restrict__ H,
                            const float* __restrict__ dinv, const float* __restrict__ bconv,
                            const int* __restrict__ batch,
                            float* __restrict__ hcat, float* __restrict__ gcnt,
                            int N, int side) {
    long long gid = (long long)blockIdx.x * 256 + threadIdx.x;
    int i = (int)(gid >> 6);
    int f = (int)(gid & 63);
    if (i < N) {
        float di = dinv[i];
        float v = agg[(size_t)i * 64 + f] + di * di * H[(size_t)i * 64 + f] + bconv[f];
        v = v > 0.0f ? v : 0.0f;   // ReLU
        int g = batch[i];
        atomicAdd(&hcat[(size_t)g * 128 + side * 64 + f], v);
        if (f == 0) atomicAdd(&gcnt[side * NGRAPHS + g], 1.0f);
    }
}

// ---------------- per-graph MLP head: [128]->64->32->16->1, sigmoid ----------------
__global__ void mlp_kernel(const float* __restrict__ hcat, const float* __restrict__ gcnt,
                           const float* __restrict__ W1, const float* __restrict__ B1,
                           const float* __restrict__ W2, const float* __restrict__ B2,
                           const float* __restrict__ W3, const float* __restrict__ B3,
                           const float* __restrict__ W4, const float* __restrict__ B4,
                           float* __restrict__ out) {
    __shared__ float bufA[128];
    __shared__ float bufB[64];
    int t = threadIdx.x;
    int g = blockIdx.x;

    // mean pool: divide accumulated sums by per-graph node counts (per side)
    {
        int side = t >> 6;
        float c = gcnt[side * NGRAPHS + g];
        c = c > 1.0f ? c : 1.0f;
        bufA[t] = hcat[(size_t)g * 128 + t] / c;
    }
    __syncthreads();
    if (t < 64) {
        float acc = B1[t];
        for (int k = 0; k < 128; ++k) acc += bufA[k] * W1[k * 64 + t];
        bufB[t] = acc > 0.0f ? acc : 0.0f;
    }
    __syncthreads();
    if (t < 32) {
        float acc = B2[t];
        for (int k = 0; k < 64; ++k) acc += bufB[k] * W2[k * 32 + t];
        bufA[t] = acc > 0.0f ? acc : 0.0f;
    }
    __syncthreads();
    if (t < 16) {
        float acc = B3[t];
        for (int k = 0; k < 32; ++k) acc += bufA[k] * W3[k * 16 + t];
        bufB[t] = acc > 0.0f ? acc : 0.0f;
    }
    __syncthreads();
    if (t == 0) {
        float acc = B4[0];
        for (int k = 0; k < 16; ++k) acc += bufB[k] * W4[k];
        out[g] = 1.0f / (1.0f + expf(-acc));
    }
}

extern "C" void kernel_launch(void* const* d_in, const int* in_sizes, int n_in,
                              void* d_out, int out_size, void* d_ws, size_t ws_size,
                              hipStream_t stream) {
    const float* x1    = (const float*)d_in[0];
    const int*   ei1   = (const int*)d_in[1];
    const int*   bat1  = (const int*)d_in[2];
    const float* x2    = (const float*)d_in[3];
    const int*   ei2   = (const int*)d_in[4];
    const int*   bat2  = (const int*)d_in[5];
    const float* Wc    = (const float*)d_in[6];
    const float* bc    = (const float*)d_in[7];
    const float* W1    = (const float*)d_in[8];
    const float* B1    = (const float*)d_in[9];
    const float* W2    = (const float*)d_in[10];
    const float* B2    = (const float*)d_in[11];
    const float* W3    = (const float*)d_in[12];
    const float* B3    = (const float*)d_in[13];
    const float* W4    = (const float*)d_in[14];
    const float* B4    = (const float*)d_in[15];

    const int N = in_sizes[0] / 32;     // 100000
    const int E = in_sizes[1] / 2;      // 1600000

    // workspace carve-out (reused across both sides)
    char* ws = (char*)d_ws;
    size_t off = 0;
    auto carve = [&](size_t bytes) -> char* {
        char* p = ws + off;
        off += (bytes + 255) & ~(size_t)255;
        return p;
    };
    float* deg  = (float*)carve((size_t)N * 4);
    float* dinv = (float*)carve((size_t)N * 4);
    float* H    = (float*)carve((size_t)N * 64 * 4);
    float* agg  = (float*)carve((size_t)N * 64 * 4);
    float* hcat = (float*)carve((size_t)NGRAPHS * 128 * 4);
    float* gcnt = (float*)carve((size_t)2 * NGRAPHS * 4);

    hipMemsetAsync(hcat, 0, (size_t)NGRAPHS * 128 * 4, stream);
    hipMemsetAsync(gcnt, 0, (size_t)2 * NGRAPHS * 4, stream);

    for (int side = 0; side < 2; ++side) {
        const float* x     = side ? x2 : x1;
        const int*   ei    = side ? ei2 : ei1;
        const int*   batch = side ? bat2 : bat1;
        const int*   src   = ei;
        const int*   dst   = ei + E;

        hipMemsetAsync(deg, 0, (size_t)N * 4, stream);
        hipMemsetAsync(agg, 0, (size_t)N * 64 * 4, stream);

        deg_kernel<<<(E + 255) / 256, 256, 0, stream>>>(dst, E, deg);
        dinv_kernel<<<(N + 255) / 256, 256, 0, stream>>>(deg, N, dinv);

        int nrt = (N + 15) / 16;
        gemm_xw_wmma<<<(nrt + 1) / 2, 64, 0, stream>>>(x, Wc, H, N);

        long long tE = (long long)E * 64;
        scatter_kernel<<<(unsigned)((tE + 255) / 256), 256, 0, stream>>>(src, dst, dinv, H, agg, E);

        long long tN = (long long)N * 64;
        pool_kernel<<<(unsigned)((tN + 255) / 256), 256, 0, stream>>>(agg, H, dinv, bc, batch,
                                                                      hcat, gcnt, N, side);
    }

    mlp_kernel<<<NGRAPHS, 128, 0, stream>>>(hcat, gcnt, W1, B1, W2, B2, W3, B3, W4, B4,
                                            (float*)d_out);
}